// RAPiDLayer_19799799234956
// MI455X (gfx1250) — compile-verified
//
#include <hip/hip_runtime.h>
#include <cstdint>

// ---------------------------------------------------------------------------
// RAPiD detection-head decode for MI455X (gfx1250).
//
// Shapes: bbox [8,3,128,128,5] f32, conf [8,3,128,128,1] f32,
//         cls [8,3,128,128,80] f32, anchors [3,2] f32.  STRIDE = 8.
//
// Roofline: ~135MB in + ~11MB out => ~6.3us floor @ 23.3 TB/s. Sigmoid is
// monotonic, so argmax(sigmoid(cls)) == argmax(cls) and max(sigmoid) ==
// sigmoid(max): one sigmoid per cell instead of 80 removes ~63M trans ops,
// keeping the kernel purely bandwidth-bound.
//
// CDNA5 path: the dominant ~126MB class-logit stream is staged into LDS with
// the Tensor Data Mover (tensor_load_to_lds / TENSORcnt). Each wave32 issues
// ONE 10KB linear descriptor for exactly the 32 rows its own lanes consume:
//   - no __syncthreads() needed (no cross-wave sharing; tensorcnt is
//     a per-wave completion guarantee),
//   - the independent box-decode work (bbox/conf loads + transcendentals +
//     xywha stores) executes BETWEEN the TDM issue and s_wait_tensorcnt 0,
//     hiding the DMA latency behind real work.
// LDS rows are consumed with ds_load_b128 using a per-lane chunk rotation
// ((j + tid%20) % 20) to cut the 8-way bank conflict of an 80-dword row
// stride down to ~2-way.
// ---------------------------------------------------------------------------

#define NB   8
#define NA   3
#define NH   128
#define NW   128
#define NCLS 80
constexpr int    NCELLS  = NB * NA * NH * NW;  // 393216
constexpr int    TILE    = 128;                // cells per block (4 waves)
constexpr int    WROWS   = 32;                 // rows per wave
constexpr float  PSTRIDE = 8.0f;

typedef unsigned int u32x4 __attribute__((ext_vector_type(4)));
typedef int          i32x4 __attribute__((ext_vector_type(4)));
typedef int          i32x8 __attribute__((ext_vector_type(8)));

#if defined(__has_builtin)
#  if __has_builtin(__builtin_amdgcn_tensor_load_to_lds) && \
      __has_builtin(__builtin_amdgcn_s_wait_tensorcnt)
#    define HAVE_TDM 1
#  endif
#endif
#ifndef HAVE_TDM
#  define HAVE_TDM 0
#endif

__device__ __forceinline__ float sigm(float x) {
  return 1.0f / (1.0f + __expf(-x));
}

__global__ __launch_bounds__(TILE) void rapid_decode(
    const float* __restrict__ bbox,
    const float* __restrict__ conf,
    const float* __restrict__ cls,
    const float* __restrict__ anchors,
    float* __restrict__ out)
{
  __shared__ float lds_cls[TILE * NCLS];   // 40 KB: 128 rows x 80 f32

  const int tid       = threadIdx.x;
  const int tile_base = blockIdx.x * TILE;
  const int wave      = tid >> 5;          // 0..3, uniform per wave32
  const int lane      = tid & 31;

#if HAVE_TDM
  // ---- Phase 1: each wave issues ONE TDM descriptor for its own 32 rows
  // ---- (32 x 80 f32 = 2560 elements = 10 KB, linear). No wait yet.
  {
    const unsigned elems = (unsigned)(WROWS * NCLS);  // 2560 f32
    const unsigned long long ga =
        (unsigned long long)(uintptr_t)(const void*)
        (cls + (size_t)(tile_base + wave * WROWS) * NCLS);
    // Low 32 bits of the generic (flat) shared address == LDS byte offset.
    const unsigned lds_off =
        (unsigned)(uintptr_t)(const void*)&lds_cls[wave * WROWS * NCLS];

    // D# group 0: count=1 (valid user descriptor), lds_addr, global_addr, type=2
    u32x4 g0;
    g0.x = 1u;                                               // count=1
    g0.y = lds_off;                                          // lds_addr (bytes)
    g0.z = (unsigned)ga;                                     // global_addr[31:0]
    g0.w = ((unsigned)(ga >> 32) & 0x1FFFFFFu) | (2u << 30); // [56:32] | type=2

    // D# group 1: data_size=4B; 1-D tensor/tile of `elems` elements.
    i32x8 g1;
    g1[0] = (int)(2u << 16);        // data_size=2 (4 bytes); no mask/pad/iter
    g1[1] = (int)(elems << 16);     // tensor_dim0 low16  -> bits[63:48]
    g1[2] = (int)(1u << 16);        // tensor_dim0 hi16=0; tensor_dim1=1
    g1[3] = (int)(elems << 16);     // tensor_dim1 hi16=0; tile_dim0=elems
    g1[4] = 0;                      // tile_dim1=0 (unused), tile_dim2=0
    g1[5] = (int)elems;             // tensor_dim0_stride low32
    g1[6] = 0;                      // stride hi / dim1_stride lo
    g1[7] = 0;

    i32x4 gz = {0, 0, 0, 0};        // groups 2/3 unused (<=2-D tensor)
#if __clang_major__ >= 23
    i32x8 gz8 = {0, 0, 0, 0, 0, 0, 0, 0};
    __builtin_amdgcn_tensor_load_to_lds(g0, g1, gz, gz, gz8, 0);
#else
    __builtin_amdgcn_tensor_load_to_lds(g0, g1, gz, gz, 0);
#endif
  }
#else
  // ---- Fallback: wave-private coalesced float4 copy global -> LDS
  // ---- (each wave copies only the rows its own lanes will read).
  {
    const float4* src = (const float4*)
        (cls + (size_t)(tile_base + wave * WROWS) * NCLS);
    float4* dst = (float4*)&lds_cls[wave * WROWS * NCLS];
    for (int i = lane; i < WROWS * NCLS / 4; i += 32) dst[i] = src[i];
  }
#endif

  // ---- Phase 2 (overlapped with the DMA): independent box-decode stream.
  const int    c  = tile_base + tid;
  const size_t b5 = (size_t)c * 5;
  const float  b0 = bbox[b5 + 0];
  const float  b1 = bbox[b5 + 1];
  const float  b2 = bbox[b5 + 2];
  const float  b3 = bbox[b5 + 3];
  const float  b4 = bbox[b5 + 4];
  const float  cf = conf[c];

  const int w = c & (NW - 1);
  const int h = (c >> 7) & (NH - 1);
  const int a = (c >> 14) % NA;               // anchor index
  const float ax = anchors[a * 2 + 0];
  const float ay = anchors[a * 2 + 1];

  const float px    = (sigm(b0) + (float)w) * PSTRIDE;
  const float py    = (sigm(b1) + (float)h) * PSTRIDE;
  const float pw    = __expf(b2) * ax;
  const float ph    = __expf(b3) * ay;
  const float pdeg  = sigm(b4) * 360.0f - 180.0f;  // (sig*2pi - pi)/pi*180
  const float sconf = sigm(cf);

  float* o = out + b5;
  o[0] = px; o[1] = py; o[2] = pw; o[3] = ph; o[4] = pdeg;

#if HAVE_TDM
  // ---- Phase 3: drain this wave's TDM op; its lanes read only rows it
  // ---- loaded itself, so no workgroup barrier is required.
  __builtin_amdgcn_s_wait_tensorcnt(0);
  asm volatile("" ::: "memory");   // keep LDS reads below the wait
#endif

  // ---- Argmax over 80 logits from LDS (b128 reads, per-lane rotation).
  const float* row = &lds_cls[tid * NCLS];
  float vmax = -3.402823466e38f;
  int   vidx = 0;
  const int rot = tid % 20;                  // 20 float4 chunks per row
#pragma unroll 5
  for (int j = 0; j < 20; ++j) {
    int jr = j + rot; if (jr >= 20) jr -= 20;
    const float4 v    = ((const float4*)row)[jr];
    const int    base = jr * 4;
    if (v.x > vmax) { vmax = v.x; vidx = base + 0; }
    if (v.y > vmax) { vmax = v.y; vidx = base + 1; }
    if (v.z > vmax) { vmax = v.z; vidx = base + 2; }
    if (v.w > vmax) { vmax = v.w; vidx = base + 3; }
  }

  // ---- Remaining outputs: [xywha][cls_idx][score] concatenated flat.
  out[(size_t)NCELLS * 5 + c] = (float)vidx;
  out[(size_t)NCELLS * 6 + c] = sconf * sigm(vmax);  // max(sigmoid)==sigmoid(max)
}

extern "C" void kernel_launch(void* const* d_in, const int* in_sizes, int n_in,
                              void* d_out, int out_size, void* d_ws, size_t ws_size,
                              hipStream_t stream) {
  (void)in_sizes; (void)n_in; (void)out_size; (void)d_ws; (void)ws_size;
  const float* bbox    = (const float*)d_in[0];
  const float* conf    = (const float*)d_in[1];
  const float* cls     = (const float*)d_in[2];
  const float* anchors = (const float*)d_in[3];
  float*       out     = (float*)d_out;

  rapid_decode<<<dim3(NCELLS / TILE), dim3(TILE), 0, stream>>>(
      bbox, conf, cls, anchors, out);
}